// FourierGatingUnit_80693845557714
// MI455X (gfx1250) — compile-verified
//
#include <hip/hip_runtime.h>
#include <cstdint>
#include <cstddef>

// FourierGatingUnit == LayerNorm(gate) -> 15-tap circular cross-correlation
// along time (depthwise, per-channel taps) -> multiply by residual half.
// Memory-bound: ~201 MB compulsory traffic -> ~8.6 us floor at 23.3 TB/s.
// Tile staging uses the CDNA5 Tensor Data Mover (TENSORcnt) instead of
// per-lane loads; res/out use non-temporal hints to preserve L2 for the
// gate rows that neighboring tiles re-read (halo overlap).

#define B_N      8
#define T_LEN    4096
#define E_FULL   1024
#define C_HALF   512
#define K_TAPS   15
#define TILE_T   16
#define ROWS     (TILE_T + K_TAPS - 1)   // 30 rows staged per block
#define THREADS  512                     // 16 wave32 waves

typedef unsigned int v4u __attribute__((ext_vector_type(4)));
typedef int          v8i __attribute__((ext_vector_type(8)));
typedef int          v4i __attribute__((ext_vector_type(4)));

__device__ __forceinline__ float wave_sum32(float v) {
    v += __shfl_xor(v, 16, 32);
    v += __shfl_xor(v, 8, 32);
    v += __shfl_xor(v, 4, 32);
    v += __shfl_xor(v, 2, 32);
    v += __shfl_xor(v, 1, 32);
    return v;
}

// Issue one TDM 2D tile load: nrows rows of C_HALF f32, global row stride
// E_FULL elements, packed contiguously at lds_byte_addr. D# packing per
// CDNA5 ISA 8.3/8.4 (group0: count/lds/global/type; group1: dims/strides).
__device__ __forceinline__ void tdm_load_2d(uint32_t lds_byte_addr,
                                            const void* gaddr,
                                            uint32_t nrows) {
    const uint64_t ga = (uint64_t)(uintptr_t)gaddr;
    v4u g0;
    g0.x = 1u;                                           // count=1 (valid user D#)
    g0.y = lds_byte_addr;                                // lds_addr
    g0.z = (uint32_t)ga;                                 // global_addr[31:0]
    g0.w = (uint32_t)((ga >> 32) & 0x01FFFFFFu)          // global_addr[56:32]
         | (2u << 30);                                   // type=2 ("image")
    v8i g1;
    g1[0] = (int)(2u << 16);                             // data_size=2 (4B); mask/pad/iter=0
    g1[1] = (int)(((uint32_t)C_HALF & 0xFFFFu) << 16);   // tensor_dim0[15:0]
    g1[2] = (int)((nrows & 0xFFFFu) << 16);              // dim0[31:16]=0 | tensor_dim1[15:0]
    g1[3] = (int)(((uint32_t)C_HALF) << 16);             // dim1[31:16]=0 | tile_dim0=512
    g1[4] = (int)(nrows & 0xFFFFu);                      // tile_dim1=nrows, tile_dim2=0
    g1[5] = (int)E_FULL;                                 // tensor_dim0_stride lo32 (elems)
    g1[6] = 0;                                           // stride hi16 | dim1_stride lo16
    g1[7] = 0;
    const v4i gz4 = {0, 0, 0, 0};                        // groups 2/3: 2D tile, unused
    const v8i gz8 = {0, 0, 0, 0, 0, 0, 0, 0};            // trailing group: unused
    __builtin_amdgcn_tensor_load_to_lds(g0, g1, gz4, gz4, gz8, 0);
}

__global__ __launch_bounds__(THREADS)
void fgu_kernel(const float* __restrict__ x,      // [B, T, E]
                const float* __restrict__ w,      // [C_HALF, K_TAPS]
                const float* __restrict__ gamma,  // [C_HALF]
                const float* __restrict__ beta,   // [C_HALF]
                float* __restrict__ out)          // [B, T, C_HALF]
{
    extern __shared__ float smem[];              // ROWS * C_HALF floats (60 KB)

    const int tid  = threadIdx.x;
    const int wave = tid >> 5;
    const int lane = tid & 31;

    const int blocks_per_b = T_LEN / TILE_T;     // 256
    const int b  = blockIdx.x / blocks_per_b;
    const int t0 = (blockIdx.x % blocks_per_b) * TILE_T;

    // ---------------- Phase A: TDM-stage raw gate rows into LDS -------------
    // One descriptor covers the rectangular part; the circular halo wrap
    // (last tile per batch only) gets a second descriptor back at t=0.
    const int nmain = (T_LEN - t0) < ROWS ? (T_LEN - t0) : ROWS;
    if (tid < 32) {                               // wave 0 drives the DMA
        const float* gp0 = x + ((size_t)b * T_LEN + t0) * E_FULL + C_HALF;
        tdm_load_2d((uint32_t)(uintptr_t)smem, gp0, (uint32_t)nmain);
        if (nmain < ROWS) {
            const float* gp1 = x + (size_t)b * T_LEN * E_FULL + C_HALF;
            tdm_load_2d((uint32_t)(uintptr_t)(smem + nmain * C_HALF),
                        gp1, (uint32_t)(ROWS - nmain));
        }
    }

    // Overlap the DMA: prefetch residual tile, preload gamma/beta and taps.
    {
        const float* rp =
            x + ((size_t)b * T_LEN + (t0 + wave)) * E_FULL + lane * 16;
        __builtin_prefetch(rp, 0, 1);
    }
    float gam[16], bet[16];
#pragma unroll
    for (int j = 0; j < 16; ++j) {
        gam[j] = gamma[lane + 32 * j];
        bet[j] = beta[lane + 32 * j];
    }
    const int c = tid;                            // channel owned in Phase C
    float wt[K_TAPS];
#pragma unroll
    for (int k = 0; k < K_TAPS; ++k) wt[k] = w[c * K_TAPS + k];

    if (tid < 32) __builtin_amdgcn_s_wait_tensorcnt(0);
    __syncthreads();

    // ---------------- Phase B: in-place LayerNorm of each staged row --------
    // One wave per row; lane holds channels {lane, lane+32, ..., lane+480}.
    for (int r = wave; r < ROWS; r += 16) {
        float* rowp = smem + r * C_HALF;
        float vals[16];
        float s = 0.f, ss = 0.f;
#pragma unroll
        for (int j = 0; j < 16; ++j) {
            const float v = rowp[lane + 32 * j];
            vals[j] = v;
            s  += v;
            ss += v * v;
        }
        s  = wave_sum32(s);
        ss = wave_sum32(ss);
        const float mu  = s * (1.0f / (float)C_HALF);
        const float var = ss * (1.0f / (float)C_HALF) - mu * mu;
        const float rs  = rsqrtf(var + 1e-5f);
#pragma unroll
        for (int j = 0; j < 16; ++j)
            rowp[lane + 32 * j] = (vals[j] - mu) * rs * gam[j] + bet[j];
    }
    __syncthreads();

    // ---------------- Phase C: 15-tap correlation + gating ------------------
    // Sliding register window over LDS rows: 1 LDS load + 15 FMAs per output.
    // res/out are touched exactly once -> non-temporal (keep L2 for gate).
    float win[K_TAPS];
#pragma unroll
    for (int k = 0; k < K_TAPS; ++k) win[k] = smem[k * C_HALF + c];

#pragma unroll
    for (int t = 0; t < TILE_T; ++t) {
        float acc = 0.f;
#pragma unroll
        for (int k = 0; k < K_TAPS; ++k) acc = fmaf(wt[k], win[k], acc);

        const size_t tg = (size_t)b * T_LEN + (size_t)(t0 + t);
        const float res = __builtin_nontemporal_load(&x[tg * E_FULL + c]);
        __builtin_nontemporal_store(res * acc, &out[tg * C_HALF + c]);

#pragma unroll
        for (int k = 0; k < K_TAPS - 1; ++k) win[k] = win[k + 1];
        if (t + 1 < TILE_T)
            win[K_TAPS - 1] = smem[(t + K_TAPS) * C_HALF + c];
    }
}

extern "C" void kernel_launch(void* const* d_in, const int* in_sizes, int n_in,
                              void* d_out, int out_size, void* d_ws, size_t ws_size,
                              hipStream_t stream) {
    (void)in_sizes; (void)n_in; (void)d_ws; (void)ws_size; (void)out_size;
    const float* x     = (const float*)d_in[0];   // [8, 4096, 1024]
    const float* w     = (const float*)d_in[1];   // [512, 15]
    const float* gamma = (const float*)d_in[2];   // [512]
    const float* beta  = (const float*)d_in[3];   // [512]
    float* out         = (float*)d_out;           // [8, 4096, 512]

    const dim3 grid(B_N * (T_LEN / TILE_T));      // 2048 blocks
    const dim3 block(THREADS);
    const size_t shmem = (size_t)ROWS * C_HALF * sizeof(float); // 61440 B

    fgu_kernel<<<grid, block, shmem, stream>>>(x, w, gamma, beta, out);
}